// AggGraphCapsuleLayer_15066745274551
// MI455X (gfx1250) — compile-verified
//
#include <hip/hip_runtime.h>

typedef __attribute__((ext_vector_type(2))) float v2f;
typedef __attribute__((ext_vector_type(8))) float v8f;

#define EPSF 1e-7f

// LDS layout (floats):
//   uS: [2][16][64][16]  = 32768  (u[node][c][r][d])
//   bS: [2][16][64]      =  2048  (routing logits)
//   cS: [2][16][64]      =  2048  (softmax coupling coeffs)
//   oS: [2][16][16]      =   512  (per-node outputs)
#define U_OFF 0
#define B_OFF 32768
#define C_OFF 34816
#define O_OFF 36864
#define SMEM_FLOATS 37376   // 149504 bytes < 320KB WGP LDS

extern "C" __global__ __launch_bounds__(256)
void capsule_fused(const float* __restrict__ x, const float* __restrict__ W,
                   float* __restrict__ out)
{
    extern __shared__ float smem[];
    float* uS = smem + U_OFF;
    float* bS = smem + B_OFF;
    float* cS = smem + C_OFF;
    float* oS = smem + O_OFF;

    const int tid  = threadIdx.x;
    const int wv   = tid >> 5;          // wave id == input capsule i (0..7)
    const int lane = tid & 31;
    const int lrow = lane & 15;         // M row (A) / N col (B,C,D) within tile
    const int khi  = (lane >> 4) << 1;  // K sub-offset: 0 for lanes 0-15, 2 for 16-31

    // zero routing logits
    for (int k = tid; k < 2 * 16 * 64; k += 256) bS[k] = 0.0f;

    // ================= Phase 1: projection via V_WMMA_F32_16X16X4_F32 ========
    // Block covers 16 consecutive xm rows = output nodes {2*blk, 2*blk+1}.
    // Wave wv computes rows[16] x W[wv](16x256): 16 N-tiles, K=16 -> 4 chained WMMAs.
    const long mbase = (long)blockIdx.x * 16;
    const float* xp  = x + (mbase + lrow) * 128 + wv * 16;

    // A: 16x16 f32 held as 4 chunks of 16x4 (2 VGPRs each).
    // Lane<16: v0=A[row][4k], v1=A[row][4k+1]; lane>=16: K+2 (ISA 32-bit A layout).
    v2f a[4];
#pragma unroll
    for (int k = 0; k < 4; ++k)
        a[k] = *(const v2f*)(xp + 4 * k + khi);

#pragma unroll
    for (int n = 0; n < 16; ++n) {      // N-tile n == output capsule c
        v8f acc = {};
#pragma unroll
        for (int k = 0; k < 4; ++k) {
            const int p = wv * 16 + 4 * k + khi;   // K rows of B, mirrored layout
            v2f bm;
            bm.x = W[(p    ) * 256 + n * 16 + lrow];
            bm.y = W[(p + 1) * 256 + n * 16 + lrow];
            acc = __builtin_amdgcn_wmma_f32_16x16x4_f32(
                false, a[k], false, bm, (short)0, acc, false, false);
        }
        // C/D layout: vgpr j -> M = j (lanes 0-15) / j+8 (lanes 16-31), N = lrow.
        // Scatter to u[node][c=n][r = nn*8 + i][d]
#pragma unroll
        for (int j = 0; j < 8; ++j) {
            const int mloc = j + ((lane >> 4) << 3);   // row 0..15
            const int nd   = mloc >> 3;                // node 0/1
            const int nn   = mloc & 7;                 // neighbour
            uS[(((nd * 16 + n) * 64) + nn * 8 + wv) * 16 + lrow] = acc[j];
        }
    }
    __syncthreads();

    // ================= Phase 2: dynamic routing (3 iters) in LDS =============
    const int node = tid >> 7;      // 128 threads per node
    const int t    = tid & 127;

    for (int it = 0; it < 3; ++it) {
        // softmax over c (axis=1) per (node, r)
        if (t < 64) {
            const int r = t;
            float e[16], mx = -3.4e38f;
#pragma unroll
            for (int c = 0; c < 16; ++c) {
                e[c] = bS[(node * 16 + c) * 64 + r];
                mx   = fmaxf(mx, e[c]);
            }
            float s = 0.0f;
#pragma unroll
            for (int c = 0; c < 16; ++c) { e[c] = __expf(e[c] - mx); s += e[c]; }
            const float inv = 1.0f / s;
#pragma unroll
            for (int c = 0; c < 16; ++c)
                cS[(node * 16 + c) * 64 + r] = e[c] * inv;
        }
        __syncthreads();

        // outputs[c][d] = sum_r coup[c][r] * u[c][r][d]  (2 (c,d) pairs/thread)
        float acc2[2];
#pragma unroll
        for (int pp = 0; pp < 2; ++pp) {
            const int idx = t + pp * 128;
            const int c = idx >> 4, d = idx & 15;
            float s = 0.0f;
            for (int r = 0; r < 64; ++r)
                s += cS[(node * 16 + c) * 64 + r]
                   * uS[((node * 16 + c) * 64 + r) * 16 + d];
            acc2[pp] = s;
        }

        if (it == 2) {
            // last iteration: raw weighted sum -> global output
#pragma unroll
            for (int pp = 0; pp < 2; ++pp) {
                const int idx = t + pp * 128;
                const long bp = (long)blockIdx.x * 2 + node;
                out[bp * 256 + idx] = acc2[pp];
            }
        } else {
#pragma unroll
            for (int pp = 0; pp < 2; ++pp)
                oS[node * 256 + (t + pp * 128)] = acc2[pp];
            __syncthreads();

            // squash along d per (node, c)
            if (t < 16) {
                const int c = t;
                float s2 = EPSF;
#pragma unroll
                for (int d = 0; d < 16; ++d) {
                    const float v = oS[(node * 16 + c) * 16 + d];
                    s2 += v * v;
                }
                const float scale = s2 / ((1.0f + s2) * sqrtf(s2));
#pragma unroll
                for (int d = 0; d < 16; ++d)
                    oS[(node * 16 + c) * 16 + d] *= scale;
            }
            __syncthreads();

            // b[c][r] += sum_d outputs[c][d] * u[c][r][d]  (8 (c,r) pairs/thread)
#pragma unroll
            for (int pp = 0; pp < 8; ++pp) {
                const int idx = t + pp * 128;
                const int c = idx >> 6, r = idx & 63;
                float s = 0.0f;
#pragma unroll
                for (int d = 0; d < 16; ++d)
                    s += oS[(node * 16 + c) * 16 + d]
                       * uS[((node * 16 + c) * 64 + r) * 16 + d];
                bS[(node * 16 + c) * 64 + r] += s;
            }
            __syncthreads();
        }
    }
}

extern "C" void kernel_launch(void* const* d_in, const int* in_sizes, int n_in,
                              void* d_out, int out_size, void* d_ws, size_t ws_size,
                              hipStream_t stream)
{
    const float* x = (const float*)d_in[0];   // (4, 32768, 8, 16) f32
    const float* W = (const float*)d_in[1];   // (8, 16, 256) f32
    float* out = (float*)d_out;               // (4, 4096, 16, 16) f32

    const int M      = in_sizes[0] / 128;     // 131072 xm rows
    const int blocks = M / 16;                // 2 output nodes per block -> 8192
    const size_t smem = SMEM_FLOATS * sizeof(float);

    capsule_fused<<<blocks, 256, smem, stream>>>(x, W, out);
}